// Co_Attention_89240830476698
// MI455X (gfx1250) — compile-verified
//
#include <hip/hip_runtime.h>
#include <math.h>

typedef __attribute__((ext_vector_type(2))) float v2f;
typedef __attribute__((ext_vector_type(8))) float v8f;

#define B_ 2
#define C_ 96
#define H_ 256
#define W_ 256
#define NN 65536
#define HEADS_ 4
#define CH_ 24

// fused pw-conv tile geometry
#define TY 8                     // output rows per block
#define TX 64                    // output cols per block
#define HY (TY + 2)              // halo rows
#define HX (TX + 2)              // halo cols
#define HCOLS (HY * HX)          // 660 halo pixels
#define HTILES ((HCOLS + 15) / 16)  // 42 column tiles of 16
#define LSTR 673                 // odd LDS channel stride >= HTILES*16

// ---------------------------------------------------------------------------
// Fused: pointwise conv (fp32 WMMA) -> LDS halo tile -> depthwise 3x3 ->
// global store + per-channel sum-of-squares (L2 norm) accumulation.
// Block = 128 threads (4 waves); each block owns 16 output channels (mtile)
// and an 8x64 spatial tile of one batch. Grid: (W/TX, H/TY, B*Mtiles).
// SAME padding: halo coords are CLAMPED so all loads are unconditional
// (no exec-mask churn in the K loop); out-of-image columns are zeroed once
// at the LDS store (a B column only feeds its own output column).
// ---------------------------------------------------------------------------
__global__ __launch_bounds__(128) void pwdw_fused_kernel(
    const float* __restrict__ Wm,    // [M x 96] pointwise weights
    const float* __restrict__ X,     // [B, 96, NN]
    const float* __restrict__ Wdw,   // [M x 9] depthwise weights
    float* __restrict__ Out,         // [B, M, NN]
    float* __restrict__ normOut,     // [B, 96] sumsq accum (atomic)
    int Mtiles) {
  __shared__ float lds[16 * LSTR];
  __shared__ float partial[16];

  const int tid   = threadIdx.x;
  const int lane  = tid & 31;
  const int wv    = tid >> 5;
  const int khalf = lane >> 4;
  const int nl    = lane & 15;

  const int x0    = blockIdx.x * TX;
  const int y0    = blockIdx.y * TY;
  const int mtile = blockIdx.z % Mtiles;
  const int b     = blockIdx.z / Mtiles;

  const float* Xb   = X + (size_t)b * C_ * NN;
  const float* Wrow = Wm + (size_t)(mtile * 16 + nl) * C_ + 2 * khalf;

  // Hoist A fragments: one 16x96 weight slab reused for every column tile.
  v2f afrag[C_ / 4];
#pragma unroll
  for (int j = 0; j < C_ / 4; ++j) {
    afrag[j].x = Wrow[4 * j];
    afrag[j].y = Wrow[4 * j + 1];
  }

  // Stage 1: WMMA pointwise conv over halo columns into LDS.
  for (int tile = wv; tile < HTILES; tile += 4) {
    const int cidx = tile * 16 + nl;
    const int ry = cidx / HX;
    const int rx = cidx - ry * HX;
    const int gy = y0 + ry - 1;
    const int gx = x0 + rx - 1;
    const bool valid = (cidx < HCOLS) && ((unsigned)gy < H_) && ((unsigned)gx < W_);
    const int gyc = gy < 0 ? 0 : (gy > H_ - 1 ? H_ - 1 : gy);
    const int gxc = gx < 0 ? 0 : (gx > W_ - 1 ? W_ - 1 : gx);
    const float* xp = Xb + ((size_t)gyc * W_ + gxc) + (size_t)(2 * khalf) * NN;
    v8f acc = {};
#pragma unroll
    for (int j = 0; j < C_ / 4; ++j) {
      v2f bv;
      bv.x = xp[0];
      bv.y = xp[NN];
      acc = __builtin_amdgcn_wmma_f32_16x16x4_f32(false, afrag[j], false, bv,
                                                  (short)0, acc, false, false);
      xp += (size_t)4 * NN;
    }
#pragma unroll
    for (int v = 0; v < 8; ++v) {
      const int mm = v + (khalf << 3);
      lds[mm * LSTR + cidx] = valid ? acc[v] : 0.f;
    }
  }

  if (tid < 16) partial[tid] = 0.f;
  __syncthreads();

  // Stage 2: depthwise 3x3 from LDS halo; coalesced global writes; norm accum.
  float nloc = 0.f;
  for (int i = 0; i < (16 * TY * TX) / 128; ++i) {  // 64 iterations
    const int oidx = i * 128 + tid;
    const int ch  = oidx >> 9;       // / (TY*TX)
    const int pix = oidx & (TY * TX - 1);
    const int oy  = pix >> 6;        // / TX
    const int ox  = pix & (TX - 1);
    const int gch = mtile * 16 + ch;
    const float* wp = Wdw + gch * 9;
    const float* l  = &lds[ch * LSTR + oy * HX + ox];
    float s = 0.f;
#pragma unroll
    for (int dy = 0; dy < 3; ++dy)
#pragma unroll
      for (int dx = 0; dx < 3; ++dx)
        s += wp[dy * 3 + dx] * l[dy * HX + dx];
    Out[(((size_t)b * (Mtiles * 16) + gch) * H_ + (y0 + oy)) * W_ + (x0 + ox)] = s;
    nloc += s * s;
    if ((i & 3) == 3) {              // channel boundary every 4 iterations
      if (gch < C_) atomicAdd(&partial[ch], nloc);  // ds_add_f32
      nloc = 0.f;
    }
  }
  __syncthreads();
  if (tid < 16) {
    const int gch = mtile * 16 + tid;
    if (gch < C_)
      __hip_atomic_fetch_add(&normOut[b * C_ + gch], partial[tid],
                             __ATOMIC_RELAXED, __HIP_MEMORY_SCOPE_AGENT);
  }
}

// ---------------------------------------------------------------------------
// Generic fp32 WMMA GEMM: Y[b] = Wm[b] (MxK) @ (X1[b] (+ X2[b])) (KxN).
// ADD2 is a compile-time flag (no inner-loop branch); pointer-stepped K loop.
// Grid: (Nn/64, M/16, B), block = 128 (4 waves, one 16x16 tile each).
// ---------------------------------------------------------------------------
template <bool ADD2>
__global__ void gemm_wmma_kernel(const float* __restrict__ Wm, long wBatchStride,
                                 const float* __restrict__ X1,
                                 const float* __restrict__ X2, long xBatchStride,
                                 float* __restrict__ Y, int M, int K, int Nn) {
  const int lane  = threadIdx.x & 31;
  const int wave  = threadIdx.x >> 5;
  const int b     = blockIdx.z;
  const int mtile = blockIdx.y;
  const int khalf = lane >> 4;
  const int nl    = lane & 15;
  const int ncol  = (blockIdx.x * 4 + wave) * 16 + nl;

  const float* wr = Wm + (size_t)b * wBatchStride + (size_t)(mtile * 16 + nl) * K +
                    2 * khalf;
  const float* xp = X1 + (size_t)b * xBatchStride + ncol + (size_t)(2 * khalf) * Nn;
  const float* xp2 =
      ADD2 ? (X2 + (size_t)b * xBatchStride + ncol + (size_t)(2 * khalf) * Nn) : nullptr;

  v8f acc = {};
  for (int k0 = 0; k0 < K; k0 += 4) {
    v2f a;
    a.x = wr[0];
    a.y = wr[1];
    wr += 4;
    v2f bv;
    bv.x = xp[0];
    bv.y = xp[Nn];
    if (ADD2) {
      bv.x += xp2[0];
      bv.y += xp2[Nn];
      xp2 += (size_t)4 * Nn;
    }
    acc = __builtin_amdgcn_wmma_f32_16x16x4_f32(false, a, false, bv, (short)0, acc,
                                                false, false);
    xp += (size_t)4 * Nn;
  }
#pragma unroll
  for (int v = 0; v < 8; ++v) {
    const int mm = v + (khalf << 3);
    Y[((size_t)b * M + mtile * 16 + mm) * (size_t)Nn + ncol] = acc[v];
  }
}

// ---------------------------------------------------------------------------
// Gram: G[b,96,96] += Q[b] (96 x NN) @ Km[b]^T ; spatial reduction chunked over
// blockIdx.x; fp32 WMMA; agent-scope float atomics. Grid: (64, 36, B), blk 32.
// ---------------------------------------------------------------------------
__global__ void gram_wmma_kernel(const float* __restrict__ Q,
                                 const float* __restrict__ Km, float* __restrict__ G,
                                 int qChanPerB, int kChanPerB, int spChunk) {
  const int lane  = threadIdx.x & 31;
  const int khalf = lane >> 4;
  const int nl    = lane & 15;
  const int mtile = blockIdx.y % 6;
  const int ntile = blockIdx.y / 6;
  const int b     = blockIdx.z;

  const float* qp = Q + ((size_t)b * qChanPerB + mtile * 16 + nl) * (size_t)NN +
                    blockIdx.x * spChunk + 2 * khalf;
  const float* kp = Km + ((size_t)b * kChanPerB + ntile * 16 + nl) * (size_t)NN +
                    blockIdx.x * spChunk + 2 * khalf;

  v8f acc = {};
  for (int k0 = 0; k0 < spChunk; k0 += 4) {
    v2f a;
    a.x = qp[0];
    a.y = qp[1];
    v2f bv;
    bv.x = kp[0];
    bv.y = kp[1];
    acc = __builtin_amdgcn_wmma_f32_16x16x4_f32(false, a, false, bv, (short)0, acc,
                                                false, false);
    qp += 4;
    kp += 4;
  }
#pragma unroll
  for (int v = 0; v < 8; ++v) {
    const int mm = v + (khalf << 3);
    float* dst = &G[((size_t)b * C_ + mtile * 16 + mm) * C_ + ntile * 16 + nl];
    __hip_atomic_fetch_add(dst, acc[v], __ATOMIC_RELAXED, __HIP_MEMORY_SCOPE_AGENT);
  }
}

// ---------------------------------------------------------------------------
// attn_co = softmax(softmax(qk_p*t) * softmax(qk_n*t)) per (b,head), 24x24.
// ---------------------------------------------------------------------------
__global__ void attn_kernel(const float* __restrict__ Gp, const float* __restrict__ Gn,
                            const float* __restrict__ nq, const float* __restrict__ nkp,
                            const float* __restrict__ nkn, const float* __restrict__ temp,
                            float* __restrict__ Aco) {
  const int b = blockIdx.x / HEADS_;
  const int h = blockIdx.x % HEADS_;
  const int c = threadIdx.x;
  if (c >= CH_) return;
  const int cg = h * CH_ + c;
  const float t = temp[h];
  const float invq = 1.0f / fmaxf(sqrtf(nq[b * C_ + cg]), 1e-12f);
  float sp[CH_], sn[CH_];
  float mx1 = -1e30f, mx2 = -1e30f;
  for (int d = 0; d < CH_; ++d) {
    const int dg = h * CH_ + d;
    const float ikp = 1.0f / fmaxf(sqrtf(nkp[b * C_ + dg]), 1e-12f);
    const float ikn = 1.0f / fmaxf(sqrtf(nkn[b * C_ + dg]), 1e-12f);
    const float vp = Gp[((size_t)b * C_ + cg) * C_ + dg] * invq * ikp * t;
    const float vn = Gn[((size_t)b * C_ + cg) * C_ + dg] * invq * ikn * t;
    sp[d] = vp;
    sn[d] = vn;
    mx1 = fmaxf(mx1, vp);
    mx2 = fmaxf(mx2, vn);
  }
  float s1 = 0.f, s2 = 0.f;
  for (int d = 0; d < CH_; ++d) {
    sp[d] = expf(sp[d] - mx1);
    s1 += sp[d];
    sn[d] = expf(sn[d] - mx2);
    s2 += sn[d];
  }
  float mx3 = -1e30f;
  for (int d = 0; d < CH_; ++d) {
    sp[d] = (sp[d] / s1) * (sn[d] / s2);
    mx3 = fmaxf(mx3, sp[d]);
  }
  float s3 = 0.f;
  for (int d = 0; d < CH_; ++d) {
    sp[d] = expf(sp[d] - mx3);
    s3 += sp[d];
  }
  for (int d = 0; d < CH_; ++d)
    Aco[(((size_t)b * HEADS_ + h) * CH_ + c) * CH_ + d] = sp[d] / s3;
}

// ---------------------------------------------------------------------------
// E[b,o,d] = sum_cl Wproj[o, h*24+cl] * Aco[b,h,cl,dl]
// ---------------------------------------------------------------------------
__global__ void buildE_kernel(const float* __restrict__ Wp, const float* __restrict__ Aco,
                              float* __restrict__ E) {
  const int idx = blockIdx.x * blockDim.x + threadIdx.x;  // < B_*C_*C_
  const int b = idx / (C_ * C_);
  const int r = idx % (C_ * C_);
  const int o = r / C_, d = r % C_;
  const int h = d / CH_, dl = d % CH_;
  float s = 0.f;
  for (int cl = 0; cl < CH_; ++cl)
    s += Wp[o * C_ + h * CH_ + cl] * Aco[(((size_t)b * HEADS_ + h) * CH_ + cl) * CH_ + dl];
  E[idx] = s;
}

__global__ void zero_kernel(float* __restrict__ p, int n) {
  const int i = blockIdx.x * blockDim.x + threadIdx.x;
  if (i < n) p[i] = 0.f;
}

// ---------------------------------------------------------------------------
extern "C" void kernel_launch(void* const* d_in, const int* in_sizes, int n_in,
                              void* d_out, int out_size, void* d_ws, size_t ws_size,
                              hipStream_t stream) {
  const float* x_curr    = (const float*)d_in[0];
  const float* x_prev    = (const float*)d_in[1];
  const float* x_next    = (const float*)d_in[2];
  const float* w_q       = (const float*)d_in[3];
  const float* w_q_dw    = (const float*)d_in[4];
  const float* w_kv_p    = (const float*)d_in[5];
  const float* w_kv_dw_p = (const float*)d_in[6];
  const float* w_kv_n    = (const float*)d_in[7];
  const float* w_kv_dw_n = (const float*)d_in[8];
  const float* w_proj    = (const float*)d_in[9];
  const float* temp      = (const float*)d_in[10];
  float* out = (float*)d_out;

  // workspace layout (floats)
  float* ws = (float*)d_ws;
  const size_t szC  = (size_t)B_ * C_ * NN;      // 12.58M
  const size_t sz2C = (size_t)B_ * 2 * C_ * NN;  // 25.17M
  float* qb  = ws;
  float* kvp = qb + szC;
  float* kvn = kvp + sz2C;
  float* Gp  = kvn + sz2C;                  // B_*96*96
  float* Gn  = Gp + B_ * C_ * C_;
  float* nq  = Gn + B_ * C_ * C_;           // B_*96 each
  float* nkp = nq + B_ * C_;
  float* nkn = nkp + B_ * C_;
  float* Aco = nkn + B_ * C_;               // B_*4*24*24
  float* Em  = Aco + B_ * HEADS_ * CH_ * CH_;  // B_*96*96

  // 0) zero atomic accumulators (Gp, Gn, nq, nkp, nkn are contiguous)
  const int nzero = 2 * B_ * C_ * C_ + 3 * B_ * C_;
  zero_kernel<<<(nzero + 255) / 256, 256, 0, stream>>>(Gp, nzero);

  // 1) fused pointwise(WMMA) + depthwise 3x3 + norm partials
  pwdw_fused_kernel<<<dim3(W_ / TX, H_ / TY, B_ * 6), 128, 0, stream>>>(
      w_q, x_curr, w_q_dw, qb, nq, 6);
  pwdw_fused_kernel<<<dim3(W_ / TX, H_ / TY, B_ * 12), 128, 0, stream>>>(
      w_kv_p, x_prev, w_kv_dw_p, kvp, nkp, 12);
  pwdw_fused_kernel<<<dim3(W_ / TX, H_ / TY, B_ * 12), 128, 0, stream>>>(
      w_kv_n, x_next, w_kv_dw_n, kvn, nkn, 12);

  // 2) Gram matrices (atomic-accumulated WMMA partials over 64 spatial chunks)
  gram_wmma_kernel<<<dim3(64, 36, B_), 32, 0, stream>>>(qb, kvp, Gp, C_, 2 * C_, NN / 64);
  gram_wmma_kernel<<<dim3(64, 36, B_), 32, 0, stream>>>(qb, kvn, Gn, C_, 2 * C_, NN / 64);

  // 3) double softmax -> attn_co
  attn_kernel<<<B_ * HEADS_, 32, 0, stream>>>(Gp, Gn, nq, nkp, nkn, temp, Aco);

  // 4) fold projection through block-diag attention: E[b] = Wproj @ BD(Aco[b])
  buildE_kernel<<<(B_ * C_ * C_) / 256, 256, 0, stream>>>(w_proj, Aco, Em);

  // 5) fused output: y[b] = E[b] @ (v_prev[b] + v_next[b])  (v = chans 96..191)
  gemm_wmma_kernel<true><<<dim3(NN / 64, C_ / 16, B_), 128, 0, stream>>>(
      Em, (long)C_ * C_, kvp + (size_t)C_ * NN, kvn + (size_t)C_ * NN,
      (long)2 * C_ * NN, out, C_, C_, NN);
}